// Runflow_45844480917597
// MI455X (gfx1250) — compile-verified
//
#include <hip/hip_runtime.h>

// ---------------------------------------------------------------------------
// MI455X (gfx1250) runflow kernel.
//  K1: contraction-warmup parallel pp-scan + pointwise q  (bandwidth bound)
//  K2: rt linear recurrence, 256-elem tiles: S = L @ Q via 4x v_wmma_f32_16x16x4_f32
//  K3: serial scan of per-block aggregates (tiny)
//  K4: carry fixup  out[t] += wz1 * wr1^(off+1) * C_blk
// ---------------------------------------------------------------------------

typedef __attribute__((ext_vector_type(2))) float v2f;
typedef __attribute__((ext_vector_type(8))) float v8f;

#define WU   128   // warm-up length (contraction 0.71^128 ~ 1e-19)
#define C1   64    // elements per thread in kernel 1
#define TILE 256   // elements per WMMA tile / per block in kernel 2

// Integer power with COMPILE-TIME trip count: unrolls to straight-line
// predicated v_cndmask/v_mul code even when n is lane-varying.
template <int BITS>
__device__ __forceinline__ float ipowf(float a, int n) {
  float r = 1.0f, p = a;
  #pragma unroll
  for (int i = 0; i < BITS; ++i) {
    if (n & 1) r *= p;
    p *= p;
    n >>= 1;
  }
  return r;
}

__device__ __forceinline__ float pp_step(float c, float p, float mr, float miss,
                                         float omev, float evc) {
  float last = (c > mr) ? (mr + (c - mr) * miss) : c;
  return (last + p) * omev - evc;
}

// ---------------- Kernel 1: pp scan (contraction warm-up) + q + shift -------
__global__ void k1_q(const float* __restrict__ x,
                     const float* f0p, const float* fcp, const float* kp,
                     const float* np,  const float* dpp, const float* evp,
                     const float* evcp,const float* mrp, const float* missp,
                     float* __restrict__ qs, int T)
{
  const float f0 = f0p[0], fc = fcp[0], kk = kp[0], nn = np[0];
  const float dp = dpp[0], ev = evp[0], evc = evcp[0];
  const float mr = mrp[0], miss = missp[0];
  const float omev = 1.0f - ev;
  const float rid  = roundf(x[8]);                 // area_id = x[0,8]
  const int   shift = (rid == 3723.0f || rid == 870.0f) ? 1 : 4;
  const float pp0 = x[7] - x[6];                   // total_prec[0] - prec[0]

  const int gid = blockIdx.x * blockDim.x + threadIdx.x;
  if (gid < shift) qs[gid] = 0.0f;                 // q_shift[0:shift) = 0

  const int s = gid * C1;
  if (s >= T) return;

  // Recover carry = pp_state[s-1] (exact for s<=WU, contraction-converged else)
  float c;
  if (s == 0) {
    c = pp0;                                       // state at t=0
  } else {
    int w = s - WU; float cc;
    if (w <= 0) { w = 0; cc = pp0; } else { cc = 0.0f; }
    for (int u = w + 1; u < s; ++u)
      cc = pp_step(cc, x[(u - 1) * 9 + 6], mr, miss, omev, evc);
    c = cc;                                        // state at s-1
  }

  const float cg = 1.49f / nn;
  #pragma unroll 4
  for (int i = 0; i < C1; ++i) {
    const int t = s + i;
    if (t > 0) c = pp_step(c, x[(t - 1) * 9 + 6], mr, miss, omev, evc);
    const float* r = x + (size_t)t * 9;
    __builtin_prefetch(r + 9 * 16, 0, 1);          // global_prefetch_b8 stream hint
    const float pp = fmaxf(c, 0.0f);               // prec_pre
    const float g = r[0], imp = r[2], ar = r[3], sl = r[4], tp = r[7];
    const float ft = (f0 - fc) * __expf(-kk * tp) + fc;
    const float qg = fmaxf(pp * 1.0e-3f * (g * ar) * (1.0f - ft), 0.0f);
    const float di = fmaxf(pp * 1.0e-3f - dp, 0.0f);
    const float qi = fmaxf(sqrtf(imp * ar) * cg * __powf(di, 5.0f / 3.0f) * sqrtf(sl), 0.0f);
    const int o = t + shift;
    if (o < T) qs[o] = qi + qg;                    // q_shift[t+shift] = q[t]
  }
}

// ---------------- Kernel 2: tile-local linear scan via WMMA -----------------
// h_t = a*h_{t-1} + b*q_t ;  out_t = wz1*h_t + wz2*q_t ; zero carry per tile.
// Tile = 16 segments x 16 steps: S = L @ Q, L lower-tri with b*a^(i-j).
__global__ void __launch_bounds__(32)
k2_scan_local(const float* __restrict__ qs, float* __restrict__ out,
              float* __restrict__ blkB,
              const float* wr1p, const float* wr2p,
              const float* wz1p, const float* wz2p, const int* epochp)
{
  __shared__ float lq[TILE];
  __shared__ float ls[16];
  __shared__ float lo[TILE];
  __shared__ float tab[32];      // zero-padded L-entry table: tab[j] = (j>=14)? b*a^(j-14) : 0
  const int l = threadIdx.x;                       // 0..31 (wave32)
  const size_t base = (size_t)blockIdx.x * TILE;

  const float a = wr1p[0], b = wr2p[0], wz1 = wz1p[0], wz2 = wz2p[0];

  #pragma unroll
  for (int i = 0; i < 8; ++i) lq[i * 32 + l] = qs[base + i * 32 + l];
  tab[l] = (l >= 14) ? b * ipowf<5>(a, l - 14) : 0.0f;   // all 32 lanes, branchless
  __syncthreads();

  if (epochp[0] <= 10) {                           // INTERVAL_EPOCH path: out = q_shift
    #pragma unroll
    for (int i = 0; i < 8; ++i) out[base + i * 32 + l] = lq[i * 32 + l];
    if (l == 0) blkB[blockIdx.x] = 0.0f;
    return;
  }

  const float a16 = ipowf<5>(a, 16);               // uniform scalar, straight-line
  const int c    = l & 15;                         // column / segment, also A-row M
  const int half = l >> 4;

  // S(16x16) = L(16x16) @ Q(16x16) as 4 chained K=4 WMMAs (ISA 7.12.2 layouts):
  //  A 16x4 f32:  lane -> M (=c), reg v + 2*half -> K ; L[m][k] = b*a^(m-k), m>=k
  //  B 4x16 f32:  lane -> N (=c), reg v + 2*half -> K ; Q[k][n] = lq[n*16+k]
  // Fragments gathered first (unconditional ds_loads, no exec branching),
  // then the 4 WMMAs issue back-to-back (D->C accumulate has no hazard).
  v2f afr[4], bfr[4];
  #pragma unroll
  for (int kb = 0; kb < 4; ++kb) {
    const int k0  = 4 * kb + 2 * half;
    const int idx = c - k0;                        // in [-14, 15]
    afr[kb].x = tab[idx + 14];                     // b*a^idx   or 0
    afr[kb].y = tab[idx + 13];                     // b*a^(idx-1) or 0
    bfr[kb].x = lq[c * 16 + k0];
    bfr[kb].y = lq[c * 16 + k0 + 1];
  }
  v8f acc = {};
  #pragma unroll
  for (int kb = 0; kb < 4; ++kb) {
    acc = __builtin_amdgcn_wmma_f32_16x16x4_f32(
        /*neg_a=*/false, afr[kb], /*neg_b=*/false, bfr[kb],
        /*c_mod=*/(short)0, acc, /*reuse_a=*/false, /*reuse_b=*/false);
  }
  // acc[v] = S[v + 8*half][c]

  if (half == 1) ls[c] = acc[7];                   // segment sums s_c = S[15][c]
  __syncthreads();

  // carry into segment c (tile-local): u_c = x_{c-1}; uniform-bound weighted
  // scan (predicated, ls[j] is a scalar read per iteration -> no divergence)
  float u = 0.0f;
  #pragma unroll
  for (int j = 0; j < 15; ++j) {
    const float sj = ls[j];
    if (j < c) u = a16 * u + sj;
  }
  if (l == 15) blkB[blockIdx.x] = a16 * u + ls[15];  // tile aggregate x_15

  float pm = ipowf<5>(a, 8 * half + 1);            // a^(M+1), branchless
  #pragma unroll
  for (int v = 0; v < 8; ++v) {
    const int M = v + 8 * half;
    const float h = acc[v] + pm * u;
    lo[c * 16 + M] = wz1 * h + wz2 * lq[c * 16 + M];
    pm *= a;
  }
  __syncthreads();
  #pragma unroll
  for (int i = 0; i < 8; ++i) out[base + i * 32 + l] = lo[i * 32 + l];
}

// ---------------- Kernel 3: scan of block aggregates ------------------------
__global__ void k3_block_scan(const float* __restrict__ blkB, float* __restrict__ blkC,
                              const float* wr1p, const int* epochp, int nb)
{
  if (epochp[0] <= 10) return;
  if (threadIdx.x != 0 || blockIdx.x != 0) return;
  const float a256 = ipowf<9>(wr1p[0], TILE);
  float c = 0.0f;
  for (int bidx = 0; bidx < nb; ++bidx) {
    blkC[bidx] = c;                                // carry INTO block bidx
    c = a256 * c + blkB[bidx];
  }
}

// ---------------- Kernel 4: carry fixup -------------------------------------
__global__ void k4_fixup(float* __restrict__ out, const float* __restrict__ blkC,
                         const float* wr1p, const float* wz1p, const int* epochp, int T)
{
  if (epochp[0] <= 10) return;
  const int t = blockIdx.x * blockDim.x + threadIdx.x;
  if (t >= T) return;
  const float cb = blkC[t >> 8];
  const int off = t & (TILE - 1);
  out[t] += wz1p[0] * ipowf<9>(wr1p[0], off + 1) * cb;
}

// ---------------------------------------------------------------------------
extern "C" void kernel_launch(void* const* d_in, const int* in_sizes, int n_in,
                              void* d_out, int out_size, void* d_ws, size_t ws_size,
                              hipStream_t stream)
{
  const float* x    = (const float*)d_in[0];
  const float* f0   = (const float*)d_in[1];
  const float* fc   = (const float*)d_in[2];
  const float* k    = (const float*)d_in[3];
  const float* n    = (const float*)d_in[4];
  const float* dp   = (const float*)d_in[5];
  const float* ev   = (const float*)d_in[6];
  const float* evc  = (const float*)d_in[7];
  const float* mr   = (const float*)d_in[8];
  const float* ms   = (const float*)d_in[9];
  const float* wr1  = (const float*)d_in[10];
  const float* wr2  = (const float*)d_in[11];
  const float* wz1  = (const float*)d_in[12];
  const float* wz2  = (const float*)d_in[13];
  const int*   epoch= (const int*)d_in[14];

  const int T  = in_sizes[0] / 9;                  // 1048576
  const int nb = (T + TILE - 1) / TILE;            // 4096 tiles

  float* qs   = (float*)d_ws;                      // T floats: q_shift
  float* blkB = qs + T;                            // nb floats: tile aggregates
  float* blkC = blkB + nb;                         // nb floats: carries

  float* out = (float*)d_out;

  const int nth1 = (T + C1 - 1) / C1;              // 16384 scan threads
  k1_q<<<dim3((nth1 + 255) / 256), dim3(256), 0, stream>>>(
      x, f0, fc, k, n, dp, ev, evc, mr, ms, qs, T);

  k2_scan_local<<<dim3(nb), dim3(32), 0, stream>>>(
      qs, out, blkB, wr1, wr2, wz1, wz2, epoch);

  k3_block_scan<<<dim3(1), dim3(32), 0, stream>>>(blkB, blkC, wr1, epoch, nb);

  k4_fixup<<<dim3((T + 255) / 256), dim3(256), 0, stream>>>(
      out, blkC, wr1, wz1, epoch, T);
}